// DisenTag_45535243272580
// MI455X (gfx1250) — compile-verified
//
#include <hip/hip_runtime.h>
#include <hip/hip_bf16.h>
#include <math.h>

#define NUM_USER 20000
#define NUM_ITEM 20000
#define NUM_TAG  10000
#define DIM      64
#define NHEADS   4
#define FACTOR_K 4
#define DIM_K    16
#define ITERATE_K 2
#define NUM_LAYER 2
#define LEAKY_ALPHA 0.2f
#define N_UT (NUM_USER + NUM_TAG)
#define N_IT (NUM_ITEM + NUM_TAG)
#define N_UI (NUM_USER + NUM_ITEM)

typedef __attribute__((ext_vector_type(2))) float v2f;
typedef __attribute__((ext_vector_type(8))) float v8f;

// ---------------------------------------------------------------- utilities
__global__ void fill_f32(float* __restrict__ p, float v, int n) {
    int i = blockIdx.x * blockDim.x + threadIdx.x;
    if (i < n) p[i] = v;
}

__global__ void copy_f32(float* __restrict__ dst, const float* __restrict__ src, int n) {
    int i = blockIdx.x * blockDim.x + threadIdx.x;
    if (i < n) dst[i] = src[i];
}

// ------------------------------------------------- dense GEMM H = X @ W (K=64)
// W (64x64, 16KB) is staged once per block into LDS; each wave then computes a
// full 16-row x 64-col output slab (4 accumulators), reusing its A fragments
// across all 4 column tiles: 64 v_wmma_f32_16x16x4_f32 per wave.
// A 16x4 f32 layout: lane = (k>=2)*16 + m, vgpr = k&1
// B 4x16  f32 layout: lane = (k>=2)*16 + n, vgpr = k&1
// C/D 16x16 f32:      lanes 0-15 -> M=v, lanes 16-31 -> M=v+8, N = lane&15
#define GEMM_WAVES 4
__global__ void gemm64_wmma(const float* __restrict__ X, const float* __restrict__ W,
                            float* __restrict__ H, int n_tiles_m) {
    __shared__ float Wl[DIM * DIM];
    for (int i = threadIdx.x; i < DIM * DIM; i += blockDim.x)
        Wl[i] = W[i];
    __syncthreads();

    int tm = blockIdx.x * GEMM_WAVES + (threadIdx.x >> 5);
    if (tm >= n_tiles_m) return;                 // uniform per wave; EXEC stays full
    int lane  = threadIdx.x & 31;
    int mn    = lane & 15;                       // row (for A) / col (for B)
    int khalf = (lane >> 4) * 2;                 // base K within the 4-wide slab

    const float* xrow = X + (size_t)(tm * 16 + mn) * DIM;
    v8f c0 = {}, c1 = {}, c2 = {}, c3 = {};
#pragma unroll
    for (int k0 = 0; k0 < DIM; k0 += 4) {
        v2f a;
        a.x = xrow[k0 + khalf];
        a.y = xrow[k0 + khalf + 1];
        const float* w0 = &Wl[(k0 + khalf) * DIM + mn];
        const float* w1 = &Wl[(k0 + khalf + 1) * DIM + mn];
        v2f b;
        b.x = w0[0];  b.y = w1[0];
        c0 = __builtin_amdgcn_wmma_f32_16x16x4_f32(false, a, false, b, (short)0, c0, false, false);
        b.x = w0[16]; b.y = w1[16];
        c1 = __builtin_amdgcn_wmma_f32_16x16x4_f32(false, a, false, b, (short)0, c1, false, false);
        b.x = w0[32]; b.y = w1[32];
        c2 = __builtin_amdgcn_wmma_f32_16x16x4_f32(false, a, false, b, (short)0, c2, false, false);
        b.x = w0[48]; b.y = w1[48];
        c3 = __builtin_amdgcn_wmma_f32_16x16x4_f32(false, a, false, b, (short)0, c3, false, false);
    }
    int rbase = tm * 16 + ((lane >> 4) * 8);
#pragma unroll
    for (int v = 0; v < 8; ++v) {
        float* hrow = H + (size_t)(rbase + v) * DIM + (lane & 15);
        hrow[0]  = c0[v];
        hrow[16] = c1[v];
        hrow[32] = c2[v];
        hrow[48] = c3[v];
    }
}

// ----------------------------------------- per-node attention scalars s1,s2
// s1[n] = h[n] . a[0:64], s2[n] = h[n] . a[64:128]   (one wave per node)
__global__ void node_scores(const float* __restrict__ H, const float* __restrict__ a,
                            float* __restrict__ s1, float* __restrict__ s2, int N) {
    int wv = (blockIdx.x * blockDim.x + threadIdx.x) >> 5;
    if (wv >= N) return;
    int lane = threadIdx.x & 31;
    const float* h = H + (size_t)wv * DIM;
    float p1 = h[lane] * a[lane] + h[lane + 32] * a[lane + 32];
    float p2 = h[lane] * a[DIM + lane] + h[lane + 32] * a[DIM + lane + 32];
#pragma unroll
    for (int off = 16; off >= 1; off >>= 1) {
        p1 += __shfl_xor(p1, off, 32);
        p2 += __shfl_xor(p2, off, 32);
    }
    if (lane == 0) { s1[wv] = p1; s2[wv] = p2; }
}

// ------------------------------------------------- GAT edge scatter (wave/edge)
__global__ void gat_edges(const int* __restrict__ edges, int E,
                          const float* __restrict__ s1, const float* __restrict__ s2,
                          const float* __restrict__ H,
                          float* __restrict__ rowsum, float* __restrict__ agg) {
    int wv = (blockIdx.x * blockDim.x + threadIdx.x) >> 5;
    if (wv >= E) return;
    int lane = threadIdx.x & 31;
    int src = edges[wv];
    int dst = edges[E + wv];
    const float* hd = H + (size_t)dst * DIM;
    __builtin_prefetch(hd + lane, 0, 0);         // global_prefetch: hide row-gather latency
    float sc = s1[src] + s2[dst];
    float lr = sc > 0.f ? sc : LEAKY_ALPHA * sc;
    float w  = __expf(-lr);
    if (lane == 0) atomicAdd(rowsum + src, w);
    float* as = agg + (size_t)src * DIM;
    atomicAdd(as + lane,      w * hd[lane]);
    atomicAdd(as + lane + 32, w * hd[lane + 32]);
}

// ------------------------------------- GAT finalize: elu(agg/rowsum), mean head
__global__ void gat_finalize(const float* __restrict__ agg, const float* __restrict__ rowsum,
                             float* __restrict__ accum, int keep) {
    int idx = blockIdx.x * blockDim.x + threadIdx.x;
    if (idx >= keep * DIM) return;
    int n = idx >> 6;
    float v = agg[idx] / rowsum[n];
    v = v > 0.f ? v : expm1f(v);
    accum[idx] += (1.0f / NHEADS) * v;
}

// ------------------------------- stage B: softmax over factors + degree scatter
__global__ void soft_rs(const float* __restrict__ Av, float* __restrict__ Af,
                        float* __restrict__ rs, const int* __restrict__ head, int E) {
    int e = blockIdx.x * blockDim.x + threadIdx.x;
    if (e >= E) return;
    float v0 = Av[e], v1 = Av[E + e], v2 = Av[2 * (size_t)E + e], v3 = Av[3 * (size_t)E + e];
    float m  = fmaxf(fmaxf(v0, v1), fmaxf(v2, v3));
    float e0 = __expf(v0 - m), e1 = __expf(v1 - m), e2 = __expf(v2 - m), e3 = __expf(v3 - m);
    float inv = 1.0f / (e0 + e1 + e2 + e3);
    e0 *= inv; e1 *= inv; e2 *= inv; e3 *= inv;
    Af[e] = e0; Af[E + e] = e1; Af[2 * (size_t)E + e] = e2; Af[3 * (size_t)E + e] = e3;
    int h = head[e];
    atomicAdd(rs + h, e0);
    atomicAdd(rs + N_UI + h, e1);
    atomicAdd(rs + 2 * N_UI + h, e2);
    atomicAdd(rs + 3 * N_UI + h, e3);
}

// ------------------ node prep: d = rsqrt(rs), fe1 = d*split, tsp = tanh(l2(split))
__global__ void node_prep(const float* __restrict__ ego, const float* __restrict__ rs,
                          float* __restrict__ dbuf, float* __restrict__ fe1,
                          float* __restrict__ tsp, int N) {
    int n = blockIdx.x * blockDim.x + threadIdx.x;
    if (n >= N) return;
#pragma unroll
    for (int i = 0; i < FACTOR_K; ++i) {
        float r = rs[i * N + n];
        float d = r > 0.f ? rsqrtf(r) : 0.f;
        dbuf[i * N + n] = d;
        float nn = 0.f;
#pragma unroll
        for (int j = 0; j < DIM_K; ++j) {
            float x = ego[(size_t)n * DIM + i * DIM_K + j];
            nn += x * x;
        }
        float inv = 1.0f / fmaxf(sqrtf(nn), 1e-12f);
#pragma unroll
        for (int j = 0; j < DIM_K; ++j) {
            float x = ego[(size_t)n * DIM + i * DIM_K + j];
            tsp[(size_t)n * DIM + i * DIM_K + j] = tanhf(x * inv);
            fe1[(size_t)n * DIM + i * DIM_K + j] = d * x;
        }
    }
}

// ------------------------- factor propagation scatter: fe2[head] += af * fe1[tail]
__global__ void factor_scatter(const int* __restrict__ head, const int* __restrict__ tail,
                               int E, const float* __restrict__ Af,
                               const float* __restrict__ fe1, float* __restrict__ fe2) {
    int wv = (blockIdx.x * blockDim.x + threadIdx.x) >> 5;
    if (wv >= E) return;
    int lane = threadIdx.x & 31;
    int h = head[wv], t = tail[wv];
    const float* ft = fe1 + (size_t)t * DIM;
    __builtin_prefetch(ft + lane, 0, 0);
    int c0 = lane, c1 = lane + 32;
    float a0 = Af[(size_t)(c0 >> 4) * E + wv];
    float a1 = Af[(size_t)(c1 >> 4) * E + wv];
    atomicAdd(fe2 + (size_t)h * DIM + c0, a0 * ft[c0]);
    atomicAdd(fe2 + (size_t)h * DIM + c1, a1 * ft[c1]);
}

// ----------------------------- node finish: fe = d*fe2, fen = l2-normalized fe
__global__ void node_fe(const float* __restrict__ fe2, const float* __restrict__ dbuf,
                        float* __restrict__ fen, int N) {
    int n = blockIdx.x * blockDim.x + threadIdx.x;
    if (n >= N) return;
#pragma unroll
    for (int i = 0; i < FACTOR_K; ++i) {
        float d = dbuf[i * N + n];
        float nn = 0.f;
        float fe[DIM_K];
#pragma unroll
        for (int j = 0; j < DIM_K; ++j) {
            float x = d * fe2[(size_t)n * DIM + i * DIM_K + j];
            fe[j] = x;
            nn += x * x;
        }
        float inv = 1.0f / fmaxf(sqrtf(nn), 1e-12f);
#pragma unroll
        for (int j = 0; j < DIM_K; ++j)
            fen[(size_t)n * DIM + i * DIM_K + j] = fe[j] * inv;
    }
}

// ------------ edge scores: s_i = <l2(fe[head])_i, tanh(l2(split[tail]))_i>, Av += s
__global__ void edge_scores(const int* __restrict__ head, const int* __restrict__ tail,
                            int E, const float* __restrict__ fen,
                            const float* __restrict__ tsp, float* __restrict__ Av) {
    int wv = (blockIdx.x * blockDim.x + threadIdx.x) >> 5;
    if (wv >= E) return;
    int lane = threadIdx.x & 31;
    int h = head[wv], t = tail[wv];
    float p0 = fen[(size_t)h * DIM + lane]      * tsp[(size_t)t * DIM + lane];       // factors 0/1
    float p1 = fen[(size_t)h * DIM + lane + 32] * tsp[(size_t)t * DIM + lane + 32];  // factors 2/3
#pragma unroll
    for (int off = 8; off >= 1; off >>= 1) {   // reduce within 16-lane factor groups
        p0 += __shfl_xor(p0, off, 32);
        p1 += __shfl_xor(p1, off, 32);
    }
    if ((lane & 15) == 0) {
        int i = lane >> 4;                     // 0 or 1
        Av[(size_t)i * E + wv]       += p0;
        Av[(size_t)(i + 2) * E + wv] += p1;
    }
}

// ------------------------------------------------ ego <- fen ; accum += fen
__global__ void update_ego(float* __restrict__ ego, float* __restrict__ acc,
                           const float* __restrict__ fen, int n) {
    int i = blockIdx.x * blockDim.x + threadIdx.x;
    if (i < n) {
        float v = fen[i];
        ego[i] = v;
        acc[i] += v;
    }
}

__global__ void final_out(float* __restrict__ out, const float* __restrict__ acc, int n) {
    int i = blockIdx.x * blockDim.x + threadIdx.x;
    if (i < n) out[i] = acc[i] * (1.0f / (NUM_LAYER + 1));
}

// =========================================================================
extern "C" void kernel_launch(void* const* d_in, const int* in_sizes, int n_in,
                              void* d_out, int out_size, void* d_ws, size_t ws_size,
                              hipStream_t stream) {
    const float* eu   = (const float*)d_in[0];
    const float* ei   = (const float*)d_in[1];
    const float* et   = (const float*)d_in[2];
    const float* ut_W = (const float*)d_in[3];
    const float* ut_a = (const float*)d_in[4];
    const float* it_W = (const float*)d_in[5];
    const float* it_a = (const float*)d_in[6];
    const int* ut_edges = (const int*)d_in[7];
    const int* it_edges = (const int*)d_in[8];
    const int* ui_edges = (const int*)d_in[9];
    const int E_ut = in_sizes[7] / 2;
    const int E_it = in_sizes[8] / 2;
    const int E_ui = in_sizes[9] / 2;
    const int* ui_head = ui_edges;
    const int* ui_tail = ui_edges + E_ui;

    // ---- workspace layout (floats)
    size_t off = 0;
    float* base = (float*)d_ws;
    auto alloc = [&](size_t n) { float* p = base + off; off += n; return p; };
    float* X      = alloc((size_t)N_UT * DIM);
    float* H      = alloc((size_t)N_UT * DIM);
    float* s1     = alloc(N_UT);
    float* s2     = alloc(N_UT);
    float* rowsum = alloc(N_UT);
    float* agg    = alloc((size_t)N_UT * DIM);
    float* ego    = alloc((size_t)N_UI * DIM);
    float* Av     = alloc((size_t)FACTOR_K * E_ui);
    float* Af     = alloc((size_t)FACTOR_K * E_ui);
    float* rs     = alloc((size_t)FACTOR_K * N_UI);
    float* dbuf   = alloc((size_t)FACTOR_K * N_UI);
    float* fe1    = alloc((size_t)N_UI * DIM);
    float* tsp    = alloc((size_t)N_UI * DIM);
    float* fe2    = alloc((size_t)N_UI * DIM);
    float* fen    = alloc((size_t)N_UI * DIM);
    float* acc    = alloc((size_t)N_UI * DIM);
    (void)ws_size;

    const int TB = 256;
    auto gr  = [](long long n, int tb) { return (int)((n + tb - 1) / tb); };
    auto grw = [](long long waves, int tb) { return (int)((waves * 32 + tb - 1) / tb); };

    // ---- zero ego (GAT mean accumulates into it)
    fill_f32<<<gr((long long)N_UI * DIM, TB), TB, 0, stream>>>(ego, 0.f, N_UI * DIM);

    // ---- two GATs: (graph 0) user-tag -> ego[0:NUM_USER], (graph 1) item-tag
    for (int g = 0; g < 2; ++g) {
        const float* e_main = g == 0 ? eu : ei;
        const float* Wg     = g == 0 ? ut_W : it_W;
        const float* ag     = g == 0 ? ut_a : it_a;
        const int*   edges  = g == 0 ? ut_edges : it_edges;
        const int    E      = g == 0 ? E_ut : E_it;
        const int    N      = g == 0 ? N_UT : N_IT;
        const int    keep   = g == 0 ? NUM_USER : NUM_ITEM;
        float* accum        = g == 0 ? ego : ego + (size_t)NUM_USER * DIM;

        // X = concat(e_main, et)
        copy_f32<<<gr((long long)keep * DIM, TB), TB, 0, stream>>>(X, e_main, keep * DIM);
        copy_f32<<<gr((long long)NUM_TAG * DIM, TB), TB, 0, stream>>>(
            X + (size_t)keep * DIM, et, NUM_TAG * DIM);

        for (int k = 0; k < NHEADS; ++k) {
            const int tiles_m = N / 16;
            gemm64_wmma<<<(tiles_m + GEMM_WAVES - 1) / GEMM_WAVES, GEMM_WAVES * 32, 0, stream>>>(
                X, Wg + (size_t)k * DIM * DIM, H, tiles_m);
            node_scores<<<grw(N, TB), TB, 0, stream>>>(H, ag + (size_t)k * 2 * DIM, s1, s2, N);
            fill_f32<<<gr(N, TB), TB, 0, stream>>>(rowsum, 0.f, N);
            fill_f32<<<gr((long long)N * DIM, TB), TB, 0, stream>>>(agg, 0.f, N * DIM);
            gat_edges<<<grw(E, TB), TB, 0, stream>>>(edges, E, s1, s2, H, rowsum, agg);
            gat_finalize<<<gr((long long)keep * DIM, TB), TB, 0, stream>>>(agg, rowsum, accum, keep);
        }
    }

    // ---- disentangled propagation on user-item graph
    copy_f32<<<gr((long long)N_UI * DIM, TB), TB, 0, stream>>>(acc, ego, N_UI * DIM);
    fill_f32<<<gr((long long)FACTOR_K * E_ui, TB), TB, 0, stream>>>(Av, 1.0f, FACTOR_K * E_ui);

    for (int layer = 0; layer < NUM_LAYER; ++layer) {
        for (int t = 0; t < ITERATE_K; ++t) {
            fill_f32<<<gr((long long)FACTOR_K * N_UI, TB), TB, 0, stream>>>(rs, 0.f, FACTOR_K * N_UI);
            soft_rs<<<gr(E_ui, TB), TB, 0, stream>>>(Av, Af, rs, ui_head, E_ui);
            node_prep<<<gr(N_UI, TB), TB, 0, stream>>>(ego, rs, dbuf, fe1, tsp, N_UI);
            fill_f32<<<gr((long long)N_UI * DIM, TB), TB, 0, stream>>>(fe2, 0.f, N_UI * DIM);
            factor_scatter<<<grw(E_ui, TB), TB, 0, stream>>>(ui_head, ui_tail, E_ui, Af, fe1, fe2);
            node_fe<<<gr(N_UI, TB), TB, 0, stream>>>(fe2, dbuf, fen, N_UI);
            edge_scores<<<grw(E_ui, TB), TB, 0, stream>>>(ui_head, ui_tail, E_ui, fen, tsp, Av);
        }
        update_ego<<<gr((long long)N_UI * DIM, TB), TB, 0, stream>>>(ego, acc, fen, N_UI * DIM);
    }

    final_out<<<gr(out_size, TB), TB, 0, stream>>>((float*)d_out, acc, out_size);
}